// NSM_90778428768737
// MI455X (gfx1250) — compile-verified
//
#include <hip/hip_runtime.h>
#include <math.h>

// Problem constants (match reference)
#define Bq     256
#define Lq     16
#define Hq     256
#define VOCABq 5000
#define Pq     4
#define NPGq   150
#define DEGq   4
#define STEPSq 4
#define OUTq   2000
#define NNODE  (Bq*NPGq)          // 38400
#define NEDGE  (Bq*NPGq*DEGq)     // 153600

typedef float v2f __attribute__((ext_vector_type(2)));
typedef float v8f __attribute__((ext_vector_type(8)));

__device__ __forceinline__ float wave_sum(float v){
#pragma unroll
  for (int o = 16; o > 0; o >>= 1) v += __shfl_xor(v, o, 32);
  return v;
}
__device__ __forceinline__ float sigf(float x){ return 1.f/(1.f+expf(-x)); }
__device__ __forceinline__ float eluf(float x){ return x > 0.f ? x : expm1f(x); }

// ---------------------------------------------------------------------------
// fp32 WMMA GEMM: C = A(MxK) @ B^T (+Cin)(+bias)(+relu),  B stored (N,K) row-major.
// EPI bit0: +bias[col]  bit1: +Cin[row,col]  bit2: relu
// One wave computes a 16x64 tile (4 x v8f accumulators, A-fragment reused 4x).
// Out-of-range N columns are *clamped* at load (never guarded in the K-loop):
// garbage flows only into output columns that are never stored, so the hot
// loop has no exec-mask manipulation and accumulators stay stationary.
// Requires M%16==0, K%4==0, even lda/ldb (all call sites satisfy this).
// ---------------------------------------------------------------------------
template<int EPI>
__global__ __launch_bounds__(256) void gemm_wmma(
    const float* __restrict__ A, int lda,
    const float* __restrict__ B, int ldb,
    float* C, int ldc,
    const float* Cin, int ldcin,
    const float* __restrict__ bias,
    int M, int N, int K)
{
  const int wave = (blockIdx.x * blockDim.x + threadIdx.x) >> 5;
  const int lane = threadIdx.x & 31;
  const int wn   = (N + 63) >> 6;                 // waves along N
  const int tm   = wave / wn;
  const int tn   = wave % wn;
  if (tm * 16 >= M) return;                       // wave-uniform exit (EXEC stays full)

  const int m16 = lane & 15;
  const int hi  = lane >> 4;                      // 0: K+0/1 , 1: K+2/3
  const float* ap = A + (size_t)(tm*16 + m16) * lda + 2*hi;
  const float* bp0, *bp1, *bp2, *bp3;
  {
    int c0 = tn*64 +  0 + m16; c0 = c0 < N ? c0 : 0;
    int c1 = tn*64 + 16 + m16; c1 = c1 < N ? c1 : 0;
    int c2 = tn*64 + 32 + m16; c2 = c2 < N ? c2 : 0;
    int c3 = tn*64 + 48 + m16; c3 = c3 < N ? c3 : 0;
    bp0 = B + (size_t)c0 * ldb + 2*hi;
    bp1 = B + (size_t)c1 * ldb + 2*hi;
    bp2 = B + (size_t)c2 * ldb + 2*hi;
    bp3 = B + (size_t)c3 * ldb + 2*hi;
  }

  v8f acc0, acc1, acc2, acc3;
#pragma unroll
  for (int r = 0; r < 8; ++r) { acc0[r]=0.f; acc1[r]=0.f; acc2[r]=0.f; acc3[r]=0.f; }

#pragma unroll 2
  for (int k = 0; k < K; k += 4) {
    v2f a  = *(const v2f*)(ap  + k);
    v2f b0 = *(const v2f*)(bp0 + k);
    v2f b1 = *(const v2f*)(bp1 + k);
    v2f b2 = *(const v2f*)(bp2 + k);
    v2f b3 = *(const v2f*)(bp3 + k);
    acc0 = __builtin_amdgcn_wmma_f32_16x16x4_f32(false, a, false, b0, (short)0, acc0, false, false);
    acc1 = __builtin_amdgcn_wmma_f32_16x16x4_f32(false, a, false, b1, (short)0, acc1, false, false);
    acc2 = __builtin_amdgcn_wmma_f32_16x16x4_f32(false, a, false, b2, (short)0, acc2, false, false);
    acc3 = __builtin_amdgcn_wmma_f32_16x16x4_f32(false, a, false, b3, (short)0, acc3, false, false);
  }

  v8f* accs[4] = {&acc0, &acc1, &acc2, &acc3};
#pragma unroll
  for (int j = 0; j < 4; ++j) {
    const int col = tn*64 + j*16 + m16;
    if (col >= N) continue;                       // store-side guard only
#pragma unroll
    for (int r = 0; r < 8; ++r) {
      const int orow = tm*16 + hi*8 + r;          // C/D layout: vgpr r -> rows r / r+8
      float v = (*accs[j])[r];
      if (EPI & 2) v += Cin[(size_t)orow * ldcin + col];
      if (EPI & 1) v += bias[col];
      if (EPI & 4) v = v > 0.f ? v : 0.f;
      C[(size_t)orow * ldc + col] = v;
    }
  }
}

// ---------------------------------------------------------------------------
// LDS-tiled transpose: in (rows x cols) -> out (cols x rows). 256 thr = 32x8.
__global__ __launch_bounds__(256) void transpose_kernel(const float* __restrict__ in,
                                                        float* __restrict__ outp,
                                                        int rows, int cols){
  __shared__ float tile[32][33];
  const int tcx = (cols + 31) / 32;
  const int bx = blockIdx.x % tcx, by = blockIdx.x / tcx;
  const int tx = threadIdx.x & 31, ty = threadIdx.x >> 5;
  for (int i = ty; i < 32; i += 8) {
    int r = by*32 + i, c = bx*32 + tx;
    if (r < rows && c < cols) tile[i][tx] = in[(size_t)r*cols + c];
  }
  __syncthreads();
  for (int i = ty; i < 32; i += 8) {
    int r = bx*32 + i, c = by*32 + tx;
    if (r < cols && c < rows) outp[(size_t)r*rows + c] = tile[tx][i];
  }
}

__global__ void fill_kernel(float* p, float v, int n){
  int i = blockIdx.x * blockDim.x + threadIdx.x;
  if (i < n) p[i] = v;
}

// out[i] = A[i,:] . vec   (one wave per 256-wide row)
__global__ void rowdot_kernel(const float* __restrict__ A, const float* __restrict__ vec,
                              float* out, int rows){
  int wave = (blockIdx.x * blockDim.x + threadIdx.x) >> 5;
  int lane = threadIdx.x & 31;
  if (wave >= rows) return;
  const float* r = A + (size_t)wave * Hq;
  float s = 0.f;
#pragma unroll
  for (int j = 0; j < 8; ++j) s += r[lane + 32*j] * vec[lane + 32*j];
  s = wave_sum(s);
  if (!lane) out[wave] = s;
}

// Row softmax over 5000 stored logits + 1 extra (default) logit, in place.
__global__ __launch_bounds__(256) void softmax_vocab(float* Pm, const float* ldef, float* pdef){
  int i = blockIdx.x, tid = threadIdx.x;
  __shared__ float red[256];
  __shared__ float s_m, s_s;
  float* row = Pm + (size_t)i * VOCABq;
  float m = -INFINITY;
  for (int v = tid; v < VOCABq; v += 256) m = fmaxf(m, row[v]);
  red[tid] = m; __syncthreads();
  for (int s = 128; s > 0; s >>= 1){ if (tid < s) red[tid] = fmaxf(red[tid], red[tid+s]); __syncthreads(); }
  if (tid == 0) s_m = fmaxf(red[0], ldef[i]);
  __syncthreads();
  float mm = s_m, acc = 0.f;
  for (int v = tid; v < VOCABq; v += 256) acc += expf(row[v] - mm);
  red[tid] = acc; __syncthreads();
  for (int s = 128; s > 0; s >>= 1){ if (tid < s) red[tid] += red[tid+s]; __syncthreads(); }
  if (tid == 0) s_s = red[0] + expf(ldef[i] - mm);
  __syncthreads();
  float inv = 1.f / s_s;
  for (int v = tid; v < VOCABq; v += 256) row[v] = expf(row[v] - mm) * inv;
  if (tid == 0) pdef[i] = expf(ldef[i] - mm) * inv;
}

// Vpad += pdef[row] * words   (default-embed contribution)
__global__ void vtag_fix(float* Vpad, const float* pdef, const float* words){
  int idx = blockIdx.x * blockDim.x + threadIdx.x;
  if (idx >= Bq*Lq*Hq) return;
  Vpad[idx] += pdef[idx >> 8] * words[idx];
}

// LSTM gate nonlinearity + masked state update
__global__ void lstm_gate(const float* __restrict__ G, const float* __restrict__ bih,
                          const float* __restrict__ bhh, float* h, float* c,
                          const int* __restrict__ lengths, int t){
  int idx = blockIdx.x * blockDim.x + threadIdx.x;
  if (idx >= Bq*Hq) return;
  int b = idx >> 8, hh = idx & 255;
  const float* g = G + (size_t)b * 4 * Hq;
  float gi = g[        hh] + bih[        hh] + bhh[        hh];
  float gf = g[  Hq  + hh] + bih[  Hq  + hh] + bhh[  Hq  + hh];
  float gg = g[2*Hq  + hh] + bih[2*Hq  + hh] + bhh[2*Hq  + hh];
  float go = g[3*Hq  + hh] + bih[3*Hq  + hh] + bhh[3*Hq  + hh];
  if (t < lengths[b]) {
    float cn = sigf(gf) * c[idx] + sigf(gi) * tanhf(gg);
    c[idx] = cn;
    h[idx] = sigf(go) * tanhf(cn);
  }
}

// Per-graph decoder attention: att = Hins.Vpad^T, masked softmax over L, R = att @ Vpad
__global__ __launch_bounds__(256) void att_kernel(const float* __restrict__ Hins,
                                                  const float* __restrict__ Vpad,
                                                  const int* __restrict__ lengths,
                                                  float* __restrict__ R){
  int b = blockIdx.x, tid = threadIdx.x, wave = tid >> 5, lane = tid & 31;
  __shared__ float sV[Lq * Hq];        // 16 KB
  __shared__ float sHn[STEPSq * Hq];   // 4 KB
  __shared__ float sA[STEPSq][Lq];
  for (int i = tid; i < Lq*Hq;     i += 256) sV[i]  = Vpad[(size_t)b * Lq*Hq + i];
  for (int i = tid; i < STEPSq*Hq; i += 256) sHn[i] = Hins[(size_t)b * STEPSq*Hq + i];
  __syncthreads();
  int len = lengths[b];
#pragma unroll
  for (int i = 0; i < 8; ++i) {                 // 8 waves x 8 dots = 64 = 4n x 16l
    int d = wave*8 + i, n = d >> 4, l = d & 15;
    float acc = 0.f;
#pragma unroll
    for (int j = 0; j < 8; ++j){ int h = lane + 32*j; acc += sHn[n*Hq + h] * sV[l*Hq + h]; }
    acc = wave_sum(acc);
    if (!lane) sA[n][l] = (l < len) ? acc : -INFINITY;
  }
  __syncthreads();
  if (tid < STEPSq) {
    float mm = -INFINITY;
    for (int l = 0; l < Lq; ++l) mm = fmaxf(mm, sA[tid][l]);
    float ss = 0.f;
    for (int l = 0; l < Lq; ++l){ float e = expf(sA[tid][l] - mm); sA[tid][l] = e; ss += e; }
    float inv = 1.f / ss;
    for (int l = 0; l < Lq; ++l) sA[tid][l] *= inv;
  }
  __syncthreads();
#pragma unroll
  for (int n = 0; n < STEPSq; ++n) {
    float acc = 0.f;
#pragma unroll
    for (int l = 0; l < Lq; ++l) acc += sA[n][l] * sV[l*Hq + tid];
    R[(size_t)b * STEPSq*Hq + n*Hq + tid] = acc;
  }
}

// prop_sim for all 4 steps up front: PS[s,b,p] = softmax_p(R[b,s,:].prop_embeds[p,:])
__global__ void propsim_kernel(const float* __restrict__ R, const float* __restrict__ PE,
                               float* __restrict__ PS){
  int s = blockIdx.x >> 8, b = blockIdx.x & 255, lane = threadIdx.x;
  const float* instr = R + ((size_t)b * STEPSq + s) * Hq;
  float d[Pq];
#pragma unroll
  for (int p = 0; p < Pq; ++p) {
    float acc = 0.f;
#pragma unroll
    for (int j = 0; j < 8; ++j){ int h = lane + 32*j; acc += instr[h] * PE[p*Hq + h]; }
    d[p] = wave_sum(acc);
  }
  if (!lane) {
    float mm = fmaxf(fmaxf(d[0], d[1]), fmaxf(d[2], d[3]));
    float e0 = expf(d[0]-mm), e1 = expf(d[1]-mm), e2 = expf(d[2]-mm), e3 = expf(d[3]-mm);
    float inv = 1.f / (e0+e1+e2+e3);
    float* o = PS + ((size_t)s * Bq + b) * Pq;
    o[0]=e0*inv; o[1]=e1*inv; o[2]=e2*inv; o[3]=e3*inv;
  }
}

// Scalarized node score: sc[n] = sum_h elu(instr_h * sum_p ps_p*T[n,p,h]) * W_state[h]
__global__ __launch_bounds__(256) void node_score(const float* __restrict__ T,
                                                  const float* __restrict__ R,
                                                  const float* __restrict__ PS,
                                                  const float* __restrict__ Wst,
                                                  float* __restrict__ sc, int s){
  int wave = (blockIdx.x * blockDim.x + threadIdx.x) >> 5;
  int lane = threadIdx.x & 31;
  if (wave >= NNODE) return;
  int g = wave / NPGq;
  const float* ps = PS + ((size_t)s * Bq + g) * Pq;
  float p0 = ps[0], p1 = ps[1], p2 = ps[2];
  const float* instr = R + ((size_t)g * STEPSq + s) * Hq;
  const float* t = T + (size_t)wave * 3 * Hq;
  float acc = 0.f;
#pragma unroll
  for (int j = 0; j < 8; ++j) {
    int h = lane + 32*j;
    float x = instr[h] * (p0*t[h] + p1*t[Hq + h] + p2*t[2*Hq + h]);
    acc += eluf(x) * Wst[h];
  }
  acc = wave_sum(acc);
  if (!lane) sc[wave] = acc;
}

// Per-graph: edge scalar reduce + both contiguous segment softmaxes + dist update
__global__ __launch_bounds__(256) void edge_update(const float* __restrict__ ET,
                                                   const float* __restrict__ R,
                                                   const float* __restrict__ PS,
                                                   const float* __restrict__ Wrel,
                                                   const int* __restrict__ esrc,
                                                   const int* __restrict__ edst,
                                                   const float* __restrict__ sc,
                                                   float* dist, int s){
  int b = blockIdx.x, tid = threadIdx.x, wave = tid >> 5, lane = tid & 31;
  __shared__ float sAgg[NPGq], sSc[NPGq], sDist[NPGq];
  __shared__ float m1, s1, m2, s2;
  if (tid < NPGq) {
    sAgg[tid]  = 0.f;
    sSc[tid]   = sc[b*NPGq + tid];
    sDist[tid] = dist[b*NPGq + tid];
  }
  __syncthreads();
  const float* instr = R + ((size_t)b * STEPSq + s) * Hq;
  const int e0 = b * NPGq * DEGq;
  for (int e = wave; e < NPGq*DEGq; e += 8) {
    const float* et = ET + (size_t)(e0 + e) * Hq;
    float acc = 0.f;
#pragma unroll
    for (int j = 0; j < 8; ++j) {
      int h = lane + 32*j;
      acc += eluf(instr[h] * et[h]) * Wrel[h];
    }
    acc = wave_sum(acc);
    if (!lane) {
      int sl = esrc[e0 + e] - b*NPGq;
      int dl = edst[e0 + e] - b*NPGq;
      atomicAdd(&sAgg[dl], sDist[sl] * acc);
    }
  }
  __syncthreads();
  if (tid == 0) {
    float mm = -INFINITY; for (int i = 0; i < NPGq; ++i) mm = fmaxf(mm, sSc[i]);
    float ss = 0.f;       for (int i = 0; i < NPGq; ++i) ss += expf(sSc[i] - mm);
    m1 = mm; s1 = ss;
    mm = -INFINITY; for (int i = 0; i < NPGq; ++i) mm = fmaxf(mm, sAgg[i]);
    ss = 0.f;       for (int i = 0; i < NPGq; ++i) ss += expf(sAgg[i] - mm);
    m2 = mm; s2 = ss;
  }
  __syncthreads();
  if (tid < NPGq) {
    float r   = PS[((size_t)s * Bq + b) * Pq + 3];
    float pst = expf(sSc[tid]  - m1) / s1;
    float prl = expf(sAgg[tid] - m2) / s2;
    dist[b*NPGq + tid] = r * prl + (1.f - r) * pst;
  }
}

// aggregated[b,h] = sum_n dist[n] * sum_{p<3} ps[p] * node_attrs[n,p,h]
__global__ __launch_bounds__(256) void final_agg(const float* __restrict__ na,
                                                 const float* __restrict__ PS,
                                                 const float* __restrict__ dist,
                                                 float* __restrict__ agg){
  int b = blockIdx.x, h = threadIdx.x;
  __shared__ float sD[NPGq];
  if (h < NPGq) sD[h] = dist[b*NPGq + h];
  __syncthreads();
  const float* ps = PS + ((size_t)3 * Bq + b) * Pq;   // prop_sim from last step
  float p0 = ps[0], p1 = ps[1], p2 = ps[2];
  float acc = 0.f;
  for (int n = 0; n < NPGq; ++n) {
    const float* base = na + (size_t)(b*NPGq + n) * 3 * Hq;
    acc += sD[n] * (p0*base[h] + p1*base[Hq + h] + p2*base[2*Hq + h]);
  }
  agg[b*Hq + h] = acc;
}

// ---------------------------------------------------------------------------
#define GEMM_GRID(M,N) dim3((unsigned)(((((size_t)(M)/16) * (((N)+63)/64)) * 32 + 255) / 256))
#define TR_GRID(R,C)   dim3((unsigned)((((R)+31)/32) * (((C)+31)/32)))

extern "C" void kernel_launch(void* const* d_in, const int* in_sizes, int n_in,
                              void* d_out, int out_size, void* d_ws, size_t ws_size,
                              hipStream_t stream) {
  const float* questions     = (const float*)d_in[0];
  const float* node_attrs    = (const float*)d_in[1];
  const float* edge_attrs    = (const float*)d_in[2];
  const float* vocab         = (const float*)d_in[3];
  const float* default_embed = (const float*)d_in[4];
  const float* W_norm        = (const float*)d_in[5];
  const float* lstm_Wih      = (const float*)d_in[6];
  const float* lstm_Whh      = (const float*)d_in[7];
  const float* lstm_bih      = (const float*)d_in[8];
  const float* lstm_bhh      = (const float*)d_in[9];
  const float* rnn_Wih       = (const float*)d_in[10];
  const float* rnn_Whh       = (const float*)d_in[11];
  const float* rnn_bih       = (const float*)d_in[12];
  const float* rnn_bhh       = (const float*)d_in[13];
  const float* prop_embeds   = (const float*)d_in[14];
  const float* Ws_property   = (const float*)d_in[15];
  const float* W_state       = (const float*)d_in[16];
  const float* W_relation    = (const float*)d_in[17];
  const float* lin_W         = (const float*)d_in[18];
  const float* lin_b         = (const float*)d_in[19];
  const int*   lengths       = (const int*)d_in[20];
  const int*   edge_src      = (const int*)d_in[23];
  const int*   edge_dst      = (const int*)d_in[24];
  float* out = (float*)d_out;

  // Workspace layout (floats)
  float* p = (float*)d_ws;
  float* WQ    = p; p += (size_t)Bq*Lq*Hq;          // 4096x256
  float* Pm    = p; p += (size_t)Bq*Lq*VOCABq;      // 4096x5000
  float* ldef  = p; p += Bq*Lq;
  float* pdef  = p; p += Bq*Lq;
  float* Vpad  = p; p += (size_t)Bq*Lq*Hq;
  float* hbuf  = p; p += Bq*Hq;
  float* cbuf  = p; p += Bq*Hq;
  float* zbuf  = p; p += Bq*Hq;
  float* Gb    = p; p += Bq*4*Hq;
  float* QW    = p; p += Bq*Hq;
  float* Hins  = p; p += Bq*STEPSq*Hq;
  float* Rb    = p; p += Bq*STEPSq*Hq;
  float* PSb   = p; p += STEPSq*Bq*Pq;
  float* distb = p; p += NNODE;
  float* scb   = p; p += NNODE;
  float* aggb  = p; p += Bq*Hq;
  float* WnT   = p; p += (size_t)Hq*Hq;             // W_norm^T  (256x256)
  float* vocT  = p; p += (size_t)Hq*VOCABq;         // vocab^T   (256x5000)
  float* Tb    = p; p += (size_t)NNODE*3*Hq;        // 118 MB
  float* ETb   = p; p += (size_t)NEDGE*Hq;          // 157 MB

  // ---- 0. one-time transposes so every GEMM takes the coalesced B^T path ----
  transpose_kernel<<<TR_GRID(Hq,Hq),256,0,stream>>>(W_norm, WnT, Hq, Hq);
  transpose_kernel<<<TR_GRID(VOCABq,Hq),256,0,stream>>>(vocab, vocT, VOCABq, Hq);

  // ---- 1. vocab normalization: Pm = softmax(words @ W_norm @ C^T), Vtag ----
  gemm_wmma<0><<<GEMM_GRID(Bq*Lq,Hq),256,0,stream>>>(
      questions,Hq, WnT,Hq, WQ,Hq, nullptr,0,nullptr, Bq*Lq,Hq,Hq);
  gemm_wmma<0><<<GEMM_GRID(Bq*Lq,VOCABq),256,0,stream>>>(
      WQ,Hq, vocab,Hq, Pm,VOCABq, nullptr,0,nullptr, Bq*Lq,VOCABq,Hq);
  rowdot_kernel<<<(Bq*Lq*32+255)/256,256,0,stream>>>(WQ, default_embed, ldef, Bq*Lq);
  softmax_vocab<<<Bq*Lq,256,0,stream>>>(Pm, ldef, pdef);
  gemm_wmma<0><<<GEMM_GRID(Bq*Lq,Hq),256,0,stream>>>(
      Pm,VOCABq, vocT,VOCABq, Vpad,Hq, nullptr,0,nullptr, Bq*Lq,Hq,VOCABq);
  vtag_fix<<<(Bq*Lq*Hq+255)/256,256,0,stream>>>(Vpad, pdef, questions);

  // ---- 2. LSTM encoder over L=16 steps ----
  fill_kernel<<<(Bq*Hq+255)/256,256,0,stream>>>(hbuf, 0.f, Bq*Hq);
  fill_kernel<<<(Bq*Hq+255)/256,256,0,stream>>>(cbuf, 0.f, Bq*Hq);
  fill_kernel<<<(Bq*Hq+255)/256,256,0,stream>>>(zbuf, 0.f, Bq*Hq);
  for (int t = 0; t < Lq; ++t) {
    gemm_wmma<0><<<GEMM_GRID(Bq,4*Hq),256,0,stream>>>(
        Vpad + t*Hq, Lq*Hq, lstm_Wih,Hq, Gb,4*Hq, nullptr,0,nullptr, Bq,4*Hq,Hq);
    gemm_wmma<2><<<GEMM_GRID(Bq,4*Hq),256,0,stream>>>(
        hbuf,Hq, lstm_Whh,Hq, Gb,4*Hq, Gb,4*Hq, nullptr, Bq,4*Hq,Hq);
    lstm_gate<<<Bq,256,0,stream>>>(Gb, lstm_bih, lstm_bhh, hbuf, cbuf, lengths, t);
  }
  // hbuf == Q from here on.

  // ---- 3. RNN decoder (Q @ Wih^T + bih hoisted out of the 4 steps) ----
  gemm_wmma<1><<<GEMM_GRID(Bq,Hq),256,0,stream>>>(
      hbuf,Hq, rnn_Wih,Hq, QW,Hq, nullptr,0, rnn_bih, Bq,Hq,Hq);
  for (int s = 0; s < STEPSq; ++s) {
    const float* Asrc = (s == 0) ? zbuf : (Hins + (s-1)*Hq);
    int lda = (s == 0) ? Hq : STEPSq*Hq;
    gemm_wmma<7><<<GEMM_GRID(Bq,Hq),256,0,stream>>>(
        Asrc,lda, rnn_Whh,Hq, Hins + s*Hq, STEPSq*Hq, QW,Hq, rnn_bhh, Bq,Hq,Hq);
  }

  // ---- 4. decoder attention -> R, prop_sim for all steps ----
  att_kernel<<<Bq,256,0,stream>>>(Hins, Vpad, lengths, Rb);
  propsim_kernel<<<STEPSq*Bq,32,0,stream>>>(Rb, prop_embeds, PSb);

  // ---- 5. loop-invariant node/edge transforms (hoisted out of graph loop) ----
  for (int pp = 0; pp < 3; ++pp)
    gemm_wmma<0><<<GEMM_GRID(NNODE,Hq),256,0,stream>>>(
        node_attrs + pp*Hq, 3*Hq, Ws_property + (size_t)pp*Hq*Hq, Hq,
        Tb + pp*Hq, 3*Hq, nullptr,0,nullptr, NNODE,Hq,Hq);
  gemm_wmma<0><<<GEMM_GRID(NEDGE,Hq),256,0,stream>>>(
      edge_attrs,Hq, Ws_property + (size_t)3*Hq*Hq, Hq,
      ETb,Hq, nullptr,0,nullptr, NEDGE,Hq,Hq);

  // ---- 6. graph propagation (scalarized; contiguous segments, no global atomics) ----
  fill_kernel<<<(NNODE+255)/256,256,0,stream>>>(distb, 1.f/(float)NPGq, NNODE);
  for (int s = 0; s < STEPSq; ++s) {
    node_score<<<(NNODE*32+255)/256,256,0,stream>>>(Tb, Rb, PSb, W_state, scb, s);
    edge_update<<<Bq,256,0,stream>>>(ETb, Rb, PSb, W_relation, edge_src, edge_dst,
                                     scb, distb, s);
  }

  // ---- 7. readout: out = [Q, aggregated] @ lin_W^T + lin_b (split, no concat) ----
  final_agg<<<Bq,256,0,stream>>>(node_attrs, PSb, distb, aggb);
  gemm_wmma<1><<<GEMM_GRID(Bq,OUTq),256,0,stream>>>(
      hbuf,Hq, lin_W, 2*Hq, out,OUTq, nullptr,0, lin_b, Bq,OUTq,Hq);
  gemm_wmma<2><<<GEMM_GRID(Bq,OUTq),256,0,stream>>>(
      aggb,Hq, lin_W + Hq, 2*Hq, out,OUTq, out,OUTq, nullptr, Bq,OUTq,Hq);
}